// SenseNovaU1Attention_9294309229022
// MI455X (gfx1250) — compile-verified
//
#include <hip/hip_runtime.h>
#include <hip/hip_bf16.h>
#include <math.h>

typedef __attribute__((ext_vector_type(2))) float v2f;
typedef __attribute__((ext_vector_type(8))) float v8f;

constexpr int S_TOK = 2048;
constexpr int HID   = 2048;
constexpr int NH    = 16;
constexpr int NKV   = 8;
constexpr int HD    = 128;
constexpr float SCALE = 0.08838834764831845f; // 128^-0.5
constexpr float NEGV  = -1e9f;

static __device__ __forceinline__ v8f wmma_f32(v2f a, v2f b, v8f c) {
  // V_WMMA_F32_16X16X4_F32 : D = A(16x4) * B(4x16) + C(16x16), full fp32
  return __builtin_amdgcn_wmma_f32_16x16x4_f32(
      /*neg_a=*/false, a, /*neg_b=*/false, b,
      /*c_mod=*/(short)0, c, /*reuse_a=*/false, /*reuse_b=*/false);
}

// GLOBAL_LOAD_ASYNC_TO_LDS_B128: per-lane 16B DMA global -> LDS, ASYNCcnt.
// VDST = LDS byte address (32-bit VGPR), VADDR = 64-bit global address, no saddr.
static __device__ __forceinline__ void async_ld16(const float* g, const float* lds) {
  unsigned lds_off = (unsigned)(uintptr_t)lds;  // low 32 bits of generic = LDS offset
  asm volatile("global_load_async_to_lds_b128 %0, %1, off"
               :: "v"(lds_off), "v"(g) : "memory");
}
static __device__ __forceinline__ void wait_async0() {
  asm volatile("s_wait_asynccnt 0x0" ::: "memory");
}

// ---------------------------------------------------------------------------
// GEMM: C[M][N] = A[M][K] @ W[N][K]^T + bias[N]   (all fp32 row-major)
// Block tile 128x128, 8 waves as 2(M) x 4(N), per-wave 64x32 (4x2 wmma tiles).
// Double-buffered async global->LDS pipeline: one barrier per K-step.
// ---------------------------------------------------------------------------
__global__ __launch_bounds__(256) void gemm_bias_wmma(
    const float* __restrict__ A, const float* __restrict__ W,
    const float* __restrict__ bias, float* __restrict__ C,
    int M, int N, int K)
{
  constexpr int BM = 128, BN = 128, BK = 16;
  __shared__ __align__(16) float As[2][BM * BK];
  __shared__ __align__(16) float Ws[2][BN * BK];

  const int tid  = threadIdx.x;
  const int wid  = tid >> 5;
  const int lane = tid & 31;
  const int bm   = blockIdx.x * BM;
  const int bn   = blockIdx.y * BN;
  const int wm   = (wid & 1) * 64;
  const int wn   = (wid >> 1) * 32;
  const int nl   = lane & 15;
  const int ksel = (lane < 16) ? 0 : 2;

  // this thread's two float4 slots within a 128x16 tile (512 float4)
  const int i0r = tid >> 2,        i0c = (tid & 3) * 4;
  const int i1r = (tid + 256) >> 2, i1c = ((tid + 256) & 3) * 4;

  v8f acc[4][2] = {};

  const int nk = K / BK;
  // preload tile 0 into buffer 0
  {
    async_ld16(A + (size_t)(bm + i0r) * K + i0c, &As[0][i0r * BK + i0c]);
    async_ld16(W + (size_t)(bn + i0r) * K + i0c, &Ws[0][i0r * BK + i0c]);
    async_ld16(A + (size_t)(bm + i1r) * K + i1c, &As[0][i1r * BK + i1c]);
    async_ld16(W + (size_t)(bn + i1r) * K + i1c, &Ws[0][i1r * BK + i1c]);
  }

  for (int t = 0; t < nk; ++t) {
    wait_async0();       // my async stores to LDS complete
    __syncthreads();     // tile t published; tile t-1 consumers all done
    if (t + 1 < nk) {    // overlap DMA of tile t+1 with compute of tile t
      const int k1 = (t + 1) * BK;
      const int b  = (t + 1) & 1;
      async_ld16(A + (size_t)(bm + i0r) * K + k1 + i0c, &As[b][i0r * BK + i0c]);
      async_ld16(W + (size_t)(bn + i0r) * K + k1 + i0c, &Ws[b][i0r * BK + i0c]);
      async_ld16(A + (size_t)(bm + i1r) * K + k1 + i1c, &As[b][i1r * BK + i1c]);
      async_ld16(W + (size_t)(bn + i1r) * K + k1 + i1c, &Ws[b][i1r * BK + i1c]);
    }
    const float* as = As[t & 1];
    const float* ws = Ws[t & 1];
    #pragma unroll
    for (int ks = 0; ks < BK; ks += 4) {
      v2f a[4], b[2];
      #pragma unroll
      for (int mt = 0; mt < 4; mt++)
        a[mt] = *(const v2f*)(as + (wm + mt * 16 + nl) * BK + ks + ksel);
      #pragma unroll
      for (int nt = 0; nt < 2; nt++)
        b[nt] = *(const v2f*)(ws + (wn + nt * 16 + nl) * BK + ks + ksel);
      #pragma unroll
      for (int mt = 0; mt < 4; mt++)
        #pragma unroll
        for (int nt = 0; nt < 2; nt++)
          acc[mt][nt] = wmma_f32(a[mt], b[nt], acc[mt][nt]);
    }
  }

  const int half8 = (lane < 16) ? 0 : 8;
  #pragma unroll
  for (int mt = 0; mt < 4; mt++) {
    #pragma unroll
    for (int nt = 0; nt < 2; nt++) {
      #pragma unroll
      for (int r = 0; r < 8; r++) {
        int row = bm + wm + mt * 16 + r + half8;
        int col = bn + wn + nt * 16 + nl;
        C[(size_t)row * N + col] = acc[mt][nt][r] + bias[col];
      }
    }
  }
}

// ---------------------------------------------------------------------------
// RMSNorm (two halves of head_dim) + 3-way RoPE + V passthrough.
// One wave per (token, unit) where unit: 0..15 q-heads, 16..23 k-heads, 24..31 v.
// Lane l owns elements l, l+32 (time half) and l+64, l+96 (hw half).
// ---------------------------------------------------------------------------
__global__ __launch_bounds__(256) void rope_norm_kernel(
    const float* __restrict__ qkv, const int* __restrict__ idx,
    const float* __restrict__ qnw,  const float* __restrict__ knw,
    const float* __restrict__ qnhw, const float* __restrict__ knhw,
    float* __restrict__ qb, float* __restrict__ kb, float* __restrict__ vb)
{
  const int tid  = threadIdx.x;
  const int wid  = tid >> 5;
  const int lane = tid & 31;
  const int task = blockIdx.x * 8 + wid;
  const int s    = task >> 5;
  const int u    = task & 31;

  if (u >= 24) {  // V: plain copy into [KV][S][D]
    int h = u - 24;
    const float4* src = (const float4*)(qkv + (size_t)s * 4096 + 3072 + h * HD);
    float4* dst = (float4*)(vb + ((size_t)h * S_TOK + s) * HD);
    dst[lane] = src[lane];
    return;
  }
  const bool isq = (u < 16);
  const int h = isq ? u : (u - 16);
  const float* src = qkv + (size_t)s * 4096 + (isq ? h * HD : 2048 + h * HD);
  float* dst = isq ? (qb + ((size_t)h * S_TOK + s) * HD)
                   : (kb + ((size_t)h * S_TOK + s) * HD);
  const float* wt  = isq ? qnw  : knw;
  const float* whw = isq ? qnhw : knhw;

  float x0 = src[lane], x1 = src[lane + 32];       // time half (64)
  float x2 = src[lane + 64], x3 = src[lane + 96];  // hw half (64)

  // RMS over the time half
  float ss = x0 * x0 + x1 * x1;
  #pragma unroll
  for (int o = 1; o < 32; o <<= 1) ss += __shfl_xor(ss, o, 32);
  float inv = rsqrtf(ss * (1.0f / 64.0f) + 1e-6f);
  x0 = x0 * inv * wt[lane];
  x1 = x1 * inv * wt[lane + 32];

  // RMS over the hw half
  float s2 = x2 * x2 + x3 * x3;
  #pragma unroll
  for (int o = 1; o < 32; o <<= 1) s2 += __shfl_xor(s2, o, 32);
  float inv2 = rsqrtf(s2 * (1.0f / 64.0f) + 1e-6f);
  x2 = x2 * inv2 * whw[lane];
  x3 = x3 * inv2 * whw[lane + 32];

  // time RoPE: inv_freq = theta_t^-(lane/32), pairs (l, l+32)
  float post = (float)idx[s];
  float invf = powf(1.0e6f, -(float)lane * (1.0f / 32.0f));
  float ang  = post * invf;
  float ct = cosf(ang), st = sinf(ang);
  float o0 = x0 * ct - x1 * st;
  float o1 = x1 * ct + x0 * st;

  // h / w RoPE: inv_freq = theta_hw^-((lane&15)/16), pairs (l, l^16)
  float posh = (float)idx[S_TOK + s];
  float posw = (float)idx[2 * S_TOK + s];
  float invfh = powf(1.0e4f, -(float)(lane & 15) * (1.0f / 16.0f));
  float ph  = __shfl_xor(x2, 16, 32);
  float rot2 = (lane < 16) ? -ph : ph;
  float angh = posh * invfh;
  float o2 = x2 * cosf(angh) + rot2 * sinf(angh);
  float pw  = __shfl_xor(x3, 16, 32);
  float rot3 = (lane < 16) ? -pw : pw;
  float angw = posw * invfh;
  float o3 = x3 * cosf(angw) + rot3 * sinf(angw);

  dst[lane] = o0; dst[lane + 32] = o1; dst[lane + 64] = o2; dst[lane + 96] = o3;
}

// ---------------------------------------------------------------------------
// Flash attention, causal, GQA (n_rep=2). Block = (q-block of 128, head).
// Each of 8 waves owns a 16-row query tile. K/V 16x128 tiles double-buffered
// in LDS via async DMA. QK^T and PV use V_WMMA_F32_16X16X4_F32.
// ---------------------------------------------------------------------------
__global__ __launch_bounds__(256) void attn_kernel(
    const float* __restrict__ qbuf, const float* __restrict__ kbuf,
    const float* __restrict__ vbuf, float* __restrict__ obuf)
{
  __shared__ __align__(16) float Kt[2][16 * HD];
  __shared__ __align__(16) float Vt[2][16 * HD];
  __shared__ __align__(16) float Pscr[8 * 16 * 16];

  const int tid   = threadIdx.x;
  const int wid   = tid >> 5;
  const int lane  = tid & 31;
  const int qblk  = blockIdx.x;
  const int head  = blockIdx.y;
  const int kvh   = head >> 1;           // jnp.repeat(k, 2, axis=1)
  const int qbase = qblk * 128 + wid * 16;
  const int nl    = lane & 15;
  const int ksel  = (lane < 16) ? 0 : 2;
  const int half8 = (lane < 16) ? 0 : 8;

  const float* kbase = kbuf + (size_t)kvh * S_TOK * HD;
  const float* vbase = vbuf + (size_t)kvh * S_TOK * HD;

  // Q fragments: A-matrix layout for 32 k-chunks of 4 over D=128
  v2f qf[32];
  const float* qrow = qbuf + ((size_t)head * S_TOK + qbase + nl) * HD;
  #pragma unroll
  for (int c = 0; c < 32; c++) qf[c] = *(const v2f*)(qrow + 4 * c + ksel);

  v8f acc[8] = {};
  float mrun[8], lrun[8];
  #pragma unroll
  for (int r = 0; r < 8; r++) { mrun[r] = -1e30f; lrun[r] = 0.0f; }

  const int kbmax = qblk * 8 + 7;
  // preload K/V tile 0 (16x128 = 512 float4; 2 per thread per matrix)
  {
    async_ld16(kbase + (size_t)tid * 4,         &Kt[0][tid * 4]);
    async_ld16(vbase + (size_t)tid * 4,         &Vt[0][tid * 4]);
    async_ld16(kbase + (size_t)(tid + 256) * 4, &Kt[0][(tid + 256) * 4]);
    async_ld16(vbase + (size_t)(tid + 256) * 4, &Vt[0][(tid + 256) * 4]);
  }

  for (int kb = 0; kb <= kbmax; kb++) {
    wait_async0();
    __syncthreads();
    if (kb + 1 <= kbmax) {
      const float* kn = kbase + (size_t)(kb + 1) * 16 * HD;
      const float* vn = vbase + (size_t)(kb + 1) * 16 * HD;
      const int b = (kb + 1) & 1;
      async_ld16(kn + (size_t)tid * 4,         &Kt[b][tid * 4]);
      async_ld16(vn + (size_t)tid * 4,         &Vt[b][tid * 4]);
      async_ld16(kn + (size_t)(tid + 256) * 4, &Kt[b][(tid + 256) * 4]);
      async_ld16(vn + (size_t)(tid + 256) * 4, &Vt[b][(tid + 256) * 4]);
    }
    const float* kt = Kt[kb & 1];
    const float* vt = Vt[kb & 1];

    if (kb * 16 <= qbase + 15) {  // wave-uniform: tile not fully masked
      // scores = Q @ K^T over D=128 : 32 chained f32 WMMAs
      v8f sc = {};
      #pragma unroll
      for (int c = 0; c < 32; c++) {
        v2f b = *(const v2f*)(kt + nl * HD + 4 * c + ksel);
        sc = wmma_f32(qf[c], b, sc);
      }
      const int kcol = kb * 16 + nl;
      #pragma unroll
      for (int r = 0; r < 8; r++) {
        float v = sc[r] * SCALE + ((kcol > qbase + r + half8) ? NEGV : 0.0f);
        float mx = v;
        mx = fmaxf(mx, __shfl_xor(mx, 1, 32));
        mx = fmaxf(mx, __shfl_xor(mx, 2, 32));
        mx = fmaxf(mx, __shfl_xor(mx, 4, 32));
        mx = fmaxf(mx, __shfl_xor(mx, 8, 32));
        float mi    = fmaxf(mrun[r], mx);
        float alpha = __expf(mrun[r] - mi);
        mrun[r] = mi;
        float p = __expf(v - mi);
        float rs = p;
        rs += __shfl_xor(rs, 1, 32);
        rs += __shfl_xor(rs, 2, 32);
        rs += __shfl_xor(rs, 4, 32);
        rs += __shfl_xor(rs, 8, 32);
        lrun[r] = lrun[r] * alpha + rs;
        sc[r] = p;
        #pragma unroll
        for (int nb = 0; nb < 8; nb++) acc[nb][r] = acc[nb][r] * alpha;
      }
      // repack P from C-layout to A-layout via per-wave LDS scratch
      float* ps = Pscr + wid * 256;
      #pragma unroll
      for (int r = 0; r < 8; r++) ps[(r + half8) * 16 + nl] = sc[r];
      #pragma unroll
      for (int c = 0; c < 4; c++) {
        v2f a = *(const v2f*)(ps + nl * 16 + 4 * c + ksel);
        #pragma unroll
        for (int nb = 0; nb < 8; nb++) {
          v2f b;
          b[0] = vt[(4 * c + ksel) * HD + nb * 16 + nl];
          b[1] = vt[(4 * c + ksel + 1) * HD + nb * 16 + nl];
          acc[nb] = wmma_f32(a, b, acc[nb]);
        }
      }
    }
  }

  // normalize and write as attn[S][H*D]
  #pragma unroll
  for (int r = 0; r < 8; r++) {
    float invl = 1.0f / lrun[r];
    int row = qbase + r + half8;
    #pragma unroll
    for (int nb = 0; nb < 8; nb++) {
      obuf[(size_t)row * (NH * HD) + head * HD + nb * 16 + nl] = acc[nb][r] * invl;
    }
  }
}

// ---------------------------------------------------------------------------
extern "C" void kernel_launch(void* const* d_in, const int* in_sizes, int n_in,
                              void* d_out, int out_size, void* d_ws, size_t ws_size,
                              hipStream_t stream) {
  (void)in_sizes; (void)n_in; (void)out_size; (void)ws_size;
  const float* hidden = (const float*)d_in[0];
  const int*   indexes= (const int*)  d_in[1];
  // d_in[2] attention_mask: pure causal, reconstructed analytically
  const float* qkv_w  = (const float*)d_in[3];
  const float* qkv_b  = (const float*)d_in[4];
  const float* o_w    = (const float*)d_in[5];
  const float* o_b    = (const float*)d_in[6];
  const float* qnw    = (const float*)d_in[7];
  const float* knw    = (const float*)d_in[8];
  const float* qnhw   = (const float*)d_in[9];
  const float* knhw   = (const float*)d_in[10];
  float* out = (float*)d_out;

  float* ws      = (float*)d_ws;
  float* qkvbuf  = ws;                                      // [S][4096]
  float* qb      = qkvbuf + (size_t)S_TOK * 4096;           // [H][S][D]
  float* kb      = qb + (size_t)NH  * S_TOK * HD;           // [KV][S][D]
  float* vb      = kb + (size_t)NKV * S_TOK * HD;           // [KV][S][D]
  float* ob      = vb + (size_t)NKV * S_TOK * HD;           // [S][H*D]

  // 1) QKV projection: [2048,2048] @ [4096,2048]^T + b
  gemm_bias_wmma<<<dim3(S_TOK / 128, 4096 / 128), 256, 0, stream>>>(
      hidden, qkv_w, qkv_b, qkvbuf, S_TOK, 4096, HID);

  // 2) RMSNorm + RoPE(t,h,w) + V reshape : S*(16+8+8) wave-tasks, 8 per block
  rope_norm_kernel<<<S_TOK * 32 / 8, 256, 0, stream>>>(
      qkvbuf, indexes, qnw, knw, qnhw, knhw, qb, kb, vb);

  // 3) causal flash attention
  attn_kernel<<<dim3(S_TOK / 128, NH), 256, 0, stream>>>(qb, kb, vb, ob);

  // 4) output projection: [2048,2048] @ [2048,2048]^T + b
  gemm_bias_wmma<<<dim3(S_TOK / 128, HID / 128), 256, 0, stream>>>(
      ob, o_w, o_b, out, S_TOK, HID, HID);
}